// StepRelu_15771119911450
// MI455X (gfx1250) — compile-verified
//
#include <hip/hip_runtime.h>

// StepRelu quantizer for MI455X (gfx1250).
// Memory-bound streaming op: 1.074 GB traffic -> ~46 us floor at 23.3 TB/s.
// Strategy: b128 vector loads/stores with non-temporal hints (stream >> 192MB L2),
// wave32-friendly 256-thread blocks, branch-free VALU math (no divide expansion).

typedef __attribute__((ext_vector_type(4))) float v4f;

__device__ __forceinline__ float step_relu_elem(float e) {
  constexpr float theta     = 0.1f;
  constexpr float inv_theta = 10.0f;        // 1.0f/0.1f rounds exactly to 10.0f in fp32
  constexpr float upper     = 0.1f * 16.0f; // theta * STEPS_NUM (exact: 16 is a power of two)
  // binned = (ceil(x/theta) - 1) * theta
  float binned = (__builtin_ceilf(e * inv_theta) - 1.0f) * theta;
  float t = (e <= upper) ? binned : e;      // x > upper -> passthrough
  return (e <= 0.0f) ? 0.0f : t;            // x <= 0   -> 0
}

__global__ __launch_bounds__(256) void StepRelu_15771119911450_kernel(
    const float* __restrict__ x, float* __restrict__ y, long long n) {
  const long long n4 = n >> 2;
  long long i = (long long)blockIdx.x * blockDim.x + threadIdx.x;
  const long long stride = (long long)gridDim.x * blockDim.x;

  const v4f* __restrict__ xv = (const v4f*)x;
  v4f* __restrict__ yv = (v4f*)y;

  for (; i < n4; i += stride) {
    v4f v = __builtin_nontemporal_load(xv + i);   // global_load_b128, th:NT
    v4f r;
#pragma unroll
    for (int k = 0; k < 4; ++k) {
      r[k] = step_relu_elem(v[k]);
    }
    __builtin_nontemporal_store(r, yv + i);       // global_store_b128, th:NT
  }

  // Tail (n % 4). For this problem n % 4 == 0; kept for generality.
  if (blockIdx.x == 0 && threadIdx.x == 0) {
    for (long long j = n4 << 2; j < n; ++j) {
      float e = x[j];
      y[j] = step_relu_elem(e);
    }
  }
}

extern "C" void kernel_launch(void* const* d_in, const int* in_sizes, int n_in,
                              void* d_out, int out_size, void* d_ws, size_t ws_size,
                              hipStream_t stream) {
  (void)n_in; (void)out_size; (void)d_ws; (void)ws_size;
  const float* x = (const float*)d_in[0];
  float* y = (float*)d_out;
  const long long n = (long long)in_sizes[0];   // 8*4096*4096 = 134,217,728

  const int threads = 256;                      // 8 wave32s per block
  long long n4 = n >> 2;
  long long want = (n4 + threads - 1) / threads;
  if (want < 1) want = 1;
  if (want > (1ll << 20)) want = (1ll << 20);   // cap; grid-stride loop covers the rest
  const int blocks = (int)want;

  StepRelu_15771119911450_kernel<<<blocks, threads, 0, stream>>>(x, y, n);
}